// PoseModule_40656160424360
// MI455X (gfx1250) — compile-verified
//
#include <hip/hip_runtime.h>
#include <math.h>

#define LGRID 128
#define LLTOT (LGRID * LGRID)

typedef __attribute__((ext_vector_type(2))) float v2f;
typedef __attribute__((ext_vector_type(8))) float v8f;

__device__ __forceinline__ float wrap_pi(float v) {
    const float PI_F   = 3.14159265358979323846f;
    const float TWO_PI = 6.28318530717958647692f;
    float w = v + PI_F;
    w -= floorf(w * (1.0f / TWO_PI)) * TWO_PI;   // mod(w, 2*pi) in [0, 2pi)
    return w - PI_F;
}

// Kernel 1: per-batch Rodrigues rotation (rows pre-reversed) + negated offsets.
__global__ void rot_prep_kernel(const float* __restrict__ rotvec_table,
                                const float* __restrict__ offsets_table,
                                const int*   __restrict__ index,
                                float* __restrict__ rot_ws,   // B*9 (reversed rows)
                                float* __restrict__ off_ws,   // B*2 (negated)
                                int B)
{
    int b = blockIdx.x * blockDim.x + threadIdx.x;
    if (b >= B) return;
    int idx = index[b];
    float vx = rotvec_table[idx * 3 + 0];
    float vy = rotvec_table[idx * 3 + 1];
    float vz = rotvec_table[idx * 3 + 2];
    float theta = sqrtf(vx * vx + vy * vy + vz * vz);
    float inv = 1.0f / (theta + 1e-6f);
    float kx = vx * inv, ky = vy * inv, kz = vz * inv;
    float s  = sinf(theta);
    float c1 = 1.0f - cosf(theta);
    float R00 = 1.0f - c1 * (ky * ky + kz * kz);
    float R01 = -s * kz + c1 * kx * ky;
    float R02 =  s * ky + c1 * kx * kz;
    float R10 =  s * kz + c1 * kx * ky;
    float R11 = 1.0f - c1 * (kx * kx + kz * kz);
    float R12 = -s * kx + c1 * ky * kz;
    float R20 = -s * ky + c1 * kx * kz;
    float R21 =  s * kx + c1 * ky * kz;
    float R22 = 1.0f - c1 * (kx * kx + ky * ky);
    float* r = rot_ws + (size_t)b * 9;
    // pts_rot takes rows reversed: row0=R2*, row1=R1*, row2=R0*
    r[0] = R20; r[1] = R21; r[2] = R22;
    r[3] = R10; r[4] = R11; r[5] = R12;
    r[6] = R00; r[7] = R01; r[8] = R02;
    off_ws[b * 2 + 0] = -offsets_table[idx * 2 + 0];
    off_ws[b * 2 + 1] = -offsets_table[idx * 2 + 1];
}

// Kernel 2: pts_rot via V_WMMA_F32_16X16X4_F32, TWO batch items per WMMA.
// D(16x16) = A(16x4) x B(4x16):
//   A rows 0..2  = reversed rotation of batch b0,
//   A rows 8..10 = reversed rotation of batch b1,
//   B rows 0..1  = grid x/y for a 16-column tile (rows 2..3 = 0).
// D layout: VGPR i holds M=i on lanes 0..15 and M=8+i on lanes 16..31, so
// d[0..2] gives b0's three rows on the low half and b1's on the high half —
// every lane stores, no EXEC masking.
__global__ __launch_bounds__(256)
void pts_rot_kernel(const float* __restrict__ rot_ws,
                    float* __restrict__ out,   // B*3*LLTOT
                    int B)
{
    const float PI_F = 3.14159265358979323846f;
    int lane  = threadIdx.x & 31;
    int wave  = threadIdx.x >> 5;
    int b0    = blockIdx.y * 2;
    int b1    = b0 + 1;
    int col0  = (blockIdx.x * 8 + wave) * 16;

    const float* r0 = rot_ws + (size_t)b0 * 9;
    const float* r1 = rot_ws + (size_t)((b1 < B) ? b1 : b0) * 9;

    int m     = lane & 15;   // A row index (halves differ in K, not M)
    int khalf = lane >> 4;   // 0: K={0,1}, 1: K={2,3}

    // A 16x4 layout: VGPR0 = {K0 | K2}, VGPR1 = {K1 | K3} across lane halves.
    const float* rp = (m < 8) ? r0 : r1;      // rows 0..7 -> b0, 8..15 -> b1
    int  mr    = m & 7;
    bool valid = (mr < 3);
    int  mc    = valid ? mr : 0;              // in-bounds load index
    float rm0 = rp[mc * 3 + 0];
    float rm1 = rp[mc * 3 + 1];
    float rm2 = rp[mc * 3 + 2];
    float a0 = valid ? ((khalf == 0) ? rm0 : rm2) : 0.0f;
    float a1 = (valid && khalf == 0) ? rm1 : 0.0f;
    v2f a; a.x = a0; a.y = a1;

    // B 4x16 layout: VGPR0 = {row K0 | row K2}, VGPR1 = {row K1 | row K3}.
    int n = lane & 15;
    int j = col0 + n;                          // flat grid column (0..16383)
    float gx = (float)((j & 127) - 64) * (PI_F / 64.0f);  // pi * g[j%128]
    float gy = (float)((j >> 7) - 64) * (PI_F / 64.0f);   // pi * g[j/128]
    v2f bm;
    bm.x = (khalf == 0) ? gx : 0.0f;           // rows 2,3 are zero
    bm.y = (khalf == 0) ? gy : 0.0f;

    v8f c = {};
    // (neg_a, A, neg_b, B, c_mod, C, reuse_a, reuse_b)
    v8f d = __builtin_amdgcn_wmma_f32_16x16x4_f32(
        false, a, false, bm, (short)0, c, false, false);

    // Low lanes carry b0 rows 0..2, high lanes carry b1 rows 0..2.
    int bsel = (lane < 16) ? b0 : b1;
    if (bsel < B) {
        size_t base = ((size_t)bsel * 3) * LLTOT + j;
        __builtin_nontemporal_store(wrap_pi(d[0]), &out[base + 0 * (size_t)LLTOT]);
        __builtin_nontemporal_store(wrap_pi(d[1]), &out[base + 1 * (size_t)LLTOT]);
        __builtin_nontemporal_store(wrap_pi(d[2]), &out[base + 2 * (size_t)LLTOT]);
    }
}

// Kernel 3: phase = exp(i*(g[col]*(-off0) + g[row]*(-off1))) as interleaved
// (re, im) 8-byte stores. One thread per pixel, non-temporal streaming writes.
__global__ __launch_bounds__(256)
void phase_kernel(const float* __restrict__ off_ws,
                  float* __restrict__ out)  // B*LLTOT complex -> 2 floats each
{
    const float PI_F = 3.14159265358979323846f;
    long long t = (long long)blockIdx.x * blockDim.x + threadIdx.x;
    int b   = (int)(t >> 14);
    int p   = (int)(t & 16383);
    int row = p >> 7;
    int col = p & 127;
    float no0 = off_ws[b * 2 + 0];
    float no1 = off_ws[b * 2 + 1];
    float gc = (float)(col - 64) * (PI_F / 64.0f);  // (2*pi/L)*(col-64)
    float gr = (float)(row - 64) * (PI_F / 64.0f);
    float ang = gc * no0 + gr * no1;
    float s, c;
    __sincosf(ang, &s, &c);
    v2f v; v.x = c; v.y = s;
    __builtin_nontemporal_store(v, (v2f*)out + ((long long)b * LLTOT + p));
}

extern "C" void kernel_launch(void* const* d_in, const int* in_sizes, int n_in,
                              void* d_out, int out_size, void* d_ws, size_t ws_size,
                              hipStream_t stream) {
    const float* rotvec  = (const float*)d_in[0];
    const float* offsets = (const float*)d_in[1];
    const int*   index   = (const int*)d_in[2];
    int B = in_sizes[2];

    float* out    = (float*)d_out;
    float* rot_ws = (float*)d_ws;
    float* off_ws = rot_ws + (size_t)B * 9;

    rot_prep_kernel<<<(B + 255) / 256, 256, 0, stream>>>(
        rotvec, offsets, index, rot_ws, off_ws, B);

    // 8 waves/block, one 16-col tile per wave, two batch items per wave.
    dim3 gridP(LLTOT / (16 * 8), (B + 1) / 2);
    pts_rot_kernel<<<gridP, 256, 0, stream>>>(rot_ws, out, B);

    long long total = (long long)B * LLTOT;
    phase_kernel<<<(unsigned)(total / 256), 256, 0, stream>>>(
        off_ws, out + (size_t)B * 3 * LLTOT);
}